// MuellerMatrixModel_29875792511407
// MI455X (gfx1250) — compile-verified
//
#include <hip/hip_runtime.h>
#include <cstdint>

#define HGT   512
#define WID   512
#define HWSZ  (HGT*WID)
#define NB    8
#define PI_F  3.14159265358979323846f

typedef float v2f __attribute__((ext_vector_type(2)));
typedef float v8f __attribute__((ext_vector_type(8)));

__device__ __forceinline__ int imin(int a, int b) { return a < b ? a : b; }
__device__ __forceinline__ int imax(int a, int b) { return a > b ? a : b; }

// Adjugate of row-major 4x4 (A * adj(A) = det(A) * I). No division needed:
// determinants cancel in the final atan2 ratio.
__device__ __forceinline__ void adj4(const float* a, float* o) {
  float s0 = a[0]*a[5]  - a[1]*a[4];
  float s1 = a[0]*a[6]  - a[2]*a[4];
  float s2 = a[0]*a[7]  - a[3]*a[4];
  float s3 = a[1]*a[6]  - a[2]*a[5];
  float s4 = a[1]*a[7]  - a[3]*a[5];
  float s5 = a[2]*a[7]  - a[3]*a[6];
  float c5 = a[10]*a[15] - a[11]*a[14];
  float c4 = a[9]*a[15]  - a[11]*a[13];
  float c3 = a[9]*a[14]  - a[10]*a[13];
  float c2 = a[8]*a[15]  - a[11]*a[12];
  float c1 = a[8]*a[14]  - a[10]*a[12];
  float c0 = a[8]*a[13]  - a[9]*a[12];
  o[0]  =  a[5]*c5 - a[6]*c4 + a[7]*c3;
  o[1]  = -a[1]*c5 + a[2]*c4 - a[3]*c3;
  o[2]  =  a[13]*s5 - a[14]*s4 + a[15]*s3;
  o[3]  = -a[9]*s5 + a[10]*s4 - a[11]*s3;
  o[4]  = -a[4]*c5 + a[6]*c2 - a[7]*c1;
  o[5]  =  a[0]*c5 - a[2]*c2 + a[3]*c1;
  o[6]  = -a[12]*s5 + a[14]*s2 - a[15]*s1;
  o[7]  =  a[8]*s5 - a[10]*s2 + a[11]*s1;
  o[8]  =  a[4]*c4 - a[5]*c2 + a[7]*c0;
  o[9]  = -a[0]*c4 + a[1]*c2 - a[3]*c0;
  o[10] =  a[12]*s4 - a[13]*s2 + a[15]*s0;
  o[11] = -a[8]*s4 + a[9]*s2 - a[11]*s0;
  o[12] = -a[4]*c3 + a[5]*c1 - a[6]*c0;
  o[13] =  a[0]*c3 - a[1]*c1 + a[2]*c0;
  o[14] = -a[12]*s3 + a[13]*s1 - a[14]*s0;
  o[15] =  a[8]*s3 - a[9]*s1 + a[10]*s0;
}

// ---------------- Pass 1: per-pixel Mueller azimuth (memory-bound stream) ---
__global__ __launch_bounds__(256) void mueller_azi_kernel(
    const float* __restrict__ x, float* __restrict__ out) {
  int p   = blockIdx.x * 256 + threadIdx.x;     // exactly NB*HWSZ threads
  int b   = p >> 18;                            // / (512*512)
  int idx = p & (HWSZ - 1);
  const float* base = x + (size_t)b * 48 * HWSZ + idx;

  float Im[16], Am[16], Wm[16];
#pragma unroll
  for (int c = 0; c < 16; ++c) Im[c] = __builtin_nontemporal_load(base + (size_t)c * HWSZ);
#pragma unroll
  for (int c = 0; c < 16; ++c) Am[c] = __builtin_nontemporal_load(base + (size_t)(16 + c) * HWSZ);
#pragma unroll
  for (int c = 0; c < 16; ++c) Wm[c] = __builtin_nontemporal_load(base + (size_t)(32 + c) * HWSZ);

  float aA[16], aW[16];
  adj4(Am, aA);
  adj4(Wm, aW);

  // T = I * adj(W)
  float T[16];
#pragma unroll
  for (int i = 0; i < 4; ++i)
#pragma unroll
    for (int j = 0; j < 4; ++j) {
      float acc = 0.f;
#pragma unroll
      for (int k = 0; k < 4; ++k) acc = fmaf(Im[i*4+k], aW[k*4+j], acc);
      T[i*4+j] = acc;
    }

  // N = adj(A) * T; only the 5 entries feeding the retardance vector.
  float N00 = 0.f, N13 = 0.f, N23 = 0.f, N31 = 0.f, N32 = 0.f;
#pragma unroll
  for (int k = 0; k < 4; ++k) {
    N00 = fmaf(aA[k],      T[k*4+0], N00);
    N13 = fmaf(aA[4 + k],  T[k*4+3], N13);
    N23 = fmaf(aA[8 + k],  T[k*4+3], N23);
    N31 = fmaf(aA[12 + k], T[k*4+1], N31);
    N32 = fmaf(aA[12 + k], T[k*4+2], N32);
  }
  // M = N / N00 (dets cancel); scale both atan2 args by N00^2 > 0.
  float g1 = (N23 - N32) * N00;
  float g2 = (N31 - N13) * N00;
  float azi = atan2f(g2, g1) * (90.0f / PI_F);   // [-90, 90]
  if (azi < 0.f) azi += 180.f;                   // mod 180 -> [0, 180)
  out[(size_t)(b * 2) * HWSZ + idx] = azi;
}

// ---------------- Pass 2: 4x4 circular-std via async-LDS + WMMA box-sums ----
// One wave owns one 16x16 output tile. Halo tile is 19x19 (pad left/top 1,
// right/bottom 2 per SAME semantics of window 4). The separable box-sum is
// D = Band * S (vertical) then Out = V * Band (horizontal) with
// V_WMMA_F32_16X16X4_F32, K tiled in chunks of 4 (5 chunks cover K=19).
__global__ __launch_bounds__(256) void circ_std_kernel(
    const float* __restrict__ outbuf_c0, float* __restrict__ out) {
  __shared__ float lds[8][1780];
  const int lane = threadIdx.x & 31;
  const int wave = threadIdx.x >> 5;
  float* azt = &lds[wave][0];     // 19x20 azimuth halo
  float* st  = azt + 380;         // 19x20 sin(2*theta)
  float* ct  = azt + 760;         // 19x20 cos(2*theta)
  float* Vs  = azt + 1140;        // 16x20 vertical sums (sin)
  float* Vc  = azt + 1460;        // 16x20 vertical sums (cos)

  const int t  = blockIdx.x * 8 + wave;   // 8192 tiles total
  const int b  = t >> 10;
  const int tt = t & 1023;
  const int h0 = (tt >> 5) << 4;
  const int w0 = (tt & 31) << 4;
  const float* azi = outbuf_c0 + (size_t)(b * 2) * HWSZ;

  // Stage halo tile straight into LDS with CDNA5 async loads (no VGPR hop).
#pragma unroll
  for (int it = 0; it < 12; ++it) {
    int e = it * 32 + lane;
    if (e < 361) {
      int r = e / 19, c = e - r * 19;
      int hh = h0 + r - 1, ww = w0 + c - 1;
      int hc = imin(imax(hh, 0), HGT - 1);
      int wc = imin(imax(ww, 0), WID - 1);
      uint64_t ga = (uint64_t)(uintptr_t)(azi + hc * WID + wc);
      uint32_t lo = (uint32_t)(uintptr_t)(azt + r * 20 + c);
      asm volatile("global_load_async_to_lds_b32 %0, %1, off"
                   :: "v"(lo), "v"(ga) : "memory");
    }
  }
  asm volatile("s_wait_asynccnt 0" ::: "memory");
  __syncthreads();

  // Doubled-angle sin/cos; zero outside the image (reduce_window pads with 0).
#pragma unroll
  for (int it = 0; it < 12; ++it) {
    int e = it * 32 + lane;
    if (e < 361) {
      int r = e / 19, c = e - r * 19;
      int hh = h0 + r - 1, ww = w0 + c - 1;
      float sv = 0.f, cv = 0.f;
      if (hh >= 0 && hh < HGT && ww >= 0 && ww < WID) {
        float ang = azt[r * 20 + c] * (PI_F / 90.0f);
        sincosf(ang, &sv, &cv);
      }
      st[r * 20 + c] = sv;
      ct[r * 20 + c] = cv;
    }
  }
  __syncthreads();

  const int m  = lane & 15;       // A-row / B-col / D-col index for this lane
  const int k0 = (lane >> 4) * 2; // K pair selected by lane half

  // Vertical pass: V[m][c] = sum_{j=m..m+3} S[j][c]; two 16-col groups
  // (LDS cols 0..15 and 3..18) cover the 19 halo columns.
#pragma unroll
  for (int grp = 0; grp < 2; ++grp) {
    const int cg = grp * 3;
    v8f accS = {}; v8f accC = {};
#pragma unroll
    for (int ch = 0; ch < 5; ++ch) {
      int j0 = ch * 4 + k0;
      v2f a;                                         // band chunk (constant)
      a.x = (j0     >= m && j0     <= m + 3) ? 1.f : 0.f;
      a.y = (j0 + 1 >= m && j0 + 1 <= m + 3) ? 1.f : 0.f;
      int j1 = imin(j0, 18), j2 = imin(j0 + 1, 18);  // j=19 has zero band col
      v2f bs, bc;                                    // data chunk from LDS
      bs.x = st[j1 * 20 + cg + m];  bs.y = st[j2 * 20 + cg + m];
      bc.x = ct[j1 * 20 + cg + m];  bc.y = ct[j2 * 20 + cg + m];
      accS = __builtin_amdgcn_wmma_f32_16x16x4_f32(false, a, false, bs, (short)0, accS, false, false);
      accC = __builtin_amdgcn_wmma_f32_16x16x4_f32(false, a, false, bc, (short)0, accC, false, false);
    }
#pragma unroll
    for (int i = 0; i < 8; ++i) {
      int M = i + ((lane >> 4) << 3);
      if (grp == 0)      { Vs[M * 20 + m] = accS[i];     Vc[M * 20 + m] = accC[i]; }
      else if (m >= 13)  { Vs[M * 20 + 3 + m] = accS[i]; Vc[M * 20 + 3 + m] = accC[i]; }
    }
  }
  __syncthreads();

  // Horizontal pass: Out[m][n] = sum_{c=n..n+3} V[m][c].
  v8f hS = {}; v8f hC = {};
#pragma unroll
  for (int ch = 0; ch < 5; ++ch) {
    int j0 = ch * 4 + k0;
    int j1 = imin(j0, 18), j2 = imin(j0 + 1, 18);
    v2f as, ac;                                      // V data as A operand
    as.x = Vs[m * 20 + j1];  as.y = Vs[m * 20 + j2];
    ac.x = Vc[m * 20 + j1];  ac.y = Vc[m * 20 + j2];
    v2f bb;                                          // band as B operand (n = lane&15 = m)
    bb.x = (j0     >= m && j0     <= m + 3) ? 1.f : 0.f;
    bb.y = (j0 + 1 >= m && j0 + 1 <= m + 3) ? 1.f : 0.f;
    hS = __builtin_amdgcn_wmma_f32_16x16x4_f32(false, as, false, bb, (short)0, hS, false, false);
    hC = __builtin_amdgcn_wmma_f32_16x16x4_f32(false, ac, false, bb, (short)0, hC, false, false);
  }

  // Epilogue: circular std with analytic border counts.
  float* ostd = out + (size_t)(b * 2 + 1) * HWSZ;
#pragma unroll
  for (int i = 0; i < 8; ++i) {
    int M = i + ((lane >> 4) << 3);
    int h = h0 + M, w = w0 + m;
    float cnt = (float)((imin(h + 2, HGT - 1) - imax(h - 1, 0) + 1) *
                        (imin(w + 2, WID - 1) - imax(w - 1, 0) + 1));
    float inv = 1.0f / cnt;
    float ms = hS[i] * inv, mc = hC[i] * inv;
    float R = sqrtf(ms * ms + mc * mc);
    R = fminf(fmaxf(R, 1e-7f), 1.0f - 1e-7f);
    float stdv = sqrtf(-2.0f * logf(R)) * (90.0f / PI_F);
    ostd[h * WID + w] = stdv;
  }
}

extern "C" void kernel_launch(void* const* d_in, const int* in_sizes, int n_in,
                              void* d_out, int out_size, void* d_ws, size_t ws_size,
                              hipStream_t stream) {
  (void)in_sizes; (void)n_in; (void)out_size; (void)d_ws; (void)ws_size;
  const float* x = (const float*)d_in[0];
  float* out = (float*)d_out;
  // Pass 1: 2,097,152 pixels, 1 thread each.
  mueller_azi_kernel<<<dim3((NB * HWSZ) / 256), 256, 0, stream>>>(x, out);
  // Pass 2: 8192 16x16 tiles, 1 wave each, 8 waves per block.
  circ_std_kernel<<<dim3(8192 / 8), 256, 0, stream>>>(out, out);
}